// GCN_25374666785385
// MI455X (gfx1250) — compile-verified
//
#include <hip/hip_runtime.h>
#include <math.h>

#define IN_DIM  256
#define HID_DIM 64
#define OUT_DIM 40
#define OUT_PAD 48   // 3 tiles of 16, cols 40..47 zero-padded

typedef __attribute__((ext_vector_type(2))) float v2f;
typedef __attribute__((ext_vector_type(8))) float v8f;

// ---------------- degree / normalization ----------------

__global__ void k_deg_init(float* dinv, int n) {
    int i = blockIdx.x * blockDim.x + threadIdx.x;
    if (i < n) dinv[i] = 1.0f;   // self-loop contributes 1 to in-degree
}

__global__ void k_deg_accum(const long long* __restrict__ ei, float* dinv, int e) {
    int i = blockIdx.x * blockDim.x + threadIdx.x;
    if (i < e) {
        int c = (int)ei[(size_t)e + i];   // col = edge_index[1][i]
        atomicAdd(&dinv[c], 1.0f);
    }
}

__global__ void k_deg_finish(float* dinv, int n) {
    int i = blockIdx.x * blockDim.x + threadIdx.x;
    if (i < n) dinv[i] = rsqrtf(dinv[i]);   // deg >= 1 always (self-loop)
}

// ---------------- GEMM1: h1 = x @ W1  (N x 256) @ (256 x 64) ----------------
// 1 block = 128 threads = 4 waves. Block b -> rows [16b,16b+16). Wave w -> cols [16w,16w+16).
// f32 WMMA 16x16x4: A lane layout (ISA 7.12.2): lane L holds M=L&15, K = 2*(L>>4)+{0,1}.
// B symmetric: lane L holds N=L&15, K = 2*(L>>4)+{0,1}.

__global__ void k_gemm1(const float* __restrict__ X, const float* __restrict__ W,
                        float* __restrict__ H, int n) {
    const int lane  = threadIdx.x & 31;
    const int wave  = threadIdx.x >> 5;          // 0..3
    const int r16   = lane & 15;
    const int khalf = (lane >> 4) << 1;          // 0 or 2
    const int mbase = blockIdx.x * 16;
    const int row   = min(mbase + r16, n - 1);   // clamp (harmless; store guarded)
    const int col   = wave * 16 + r16;

    const float* xa = X + (size_t)row * IN_DIM + khalf;   // A: (row, khalf + k)
    const float* wb = W + (size_t)khalf * HID_DIM + col;  // B: (khalf + k, col)

    v8f c = {};
#pragma unroll 8
    for (int k = 0; k < IN_DIM; k += 4) {
        v2f a, b;
        a.x = xa[k];
        a.y = xa[k + 1];
        b.x = wb[(size_t)k * HID_DIM];
        b.y = wb[(size_t)k * HID_DIM + HID_DIM];
        c = __builtin_amdgcn_wmma_f32_16x16x4_f32(false, a, false, b,
                                                  (short)0, c, false, false);
    }

    // D layout: VGPR r -> M = r + 8*(lane>>4), N = lane&15 (within tile)
    const int mrow = mbase + ((lane >> 4) << 3);
#pragma unroll
    for (int r = 0; r < 8; ++r) {
        int m = mrow + r;
        if (m < n) H[(size_t)m * HID_DIM + col] = c[r];
    }
}

// ---------------- aggregation layer 1 (HID_DIM features) ----------------

__global__ void k_agg1_init(const float* __restrict__ h1, const float* __restrict__ dinv,
                            float* __restrict__ agg, int total) {
    int t = blockIdx.x * blockDim.x + threadIdx.x;
    if (t < total) {
        int i = t >> 6;                 // node
        float d = dinv[i];
        agg[t] = d * d * h1[t];         // self-loop term
    }
}

__global__ void k_agg1_edges(const long long* __restrict__ ei, const float* __restrict__ dinv,
                             const float* __restrict__ h1, float* __restrict__ agg, int e) {
    int t = blockIdx.x * blockDim.x + threadIdx.x;
    int total = e << 6;                 // e * 64
    if (t >= total) return;
    int eg = t >> 6, f = t & 63;
    int r = (int)ei[eg];
    int c = (int)ei[(size_t)e + eg];
    float nrm = dinv[r] * dinv[c];
    atomicAdd(&agg[(size_t)c * HID_DIM + f], nrm * h1[(size_t)r * HID_DIM + f]);
}

__global__ void k_selu_bias(float* __restrict__ agg, const float* __restrict__ b1, int total) {
    int t = blockIdx.x * blockDim.x + threadIdx.x;
    if (t < total) {
        const float alpha = 1.6732632423543772f;
        const float scale = 1.0507009873554805f;
        float x = agg[t] + b1[t & 63];
        agg[t] = scale * (x > 0.0f ? x : alpha * (__expf(x) - 1.0f));
    }
}

// ---------------- GEMM2: h2 = h @ W2  (N x 64) @ (64 x 40, padded to 48) ----------------
// 96 threads = 3 waves; wave w -> cols [16w,16w+16); W2 staged zero-padded in LDS.

__global__ void k_gemm2(const float* __restrict__ A, const float* __restrict__ W2,
                        float* __restrict__ H2, int n) {
    __shared__ float ws[HID_DIM * OUT_PAD];
    for (int idx = threadIdx.x; idx < HID_DIM * OUT_PAD; idx += blockDim.x) {
        int k = idx / OUT_PAD, nn = idx - k * OUT_PAD;
        ws[idx] = (nn < OUT_DIM) ? W2[k * OUT_DIM + nn] : 0.0f;
    }
    __syncthreads();

    const int lane  = threadIdx.x & 31;
    const int wave  = threadIdx.x >> 5;          // 0..2
    const int r16   = lane & 15;
    const int khalf = (lane >> 4) << 1;
    const int mbase = blockIdx.x * 16;
    const int row   = min(mbase + r16, n - 1);
    const int col   = wave * 16 + r16;

    const float* aa = A + (size_t)row * HID_DIM + khalf;
    const float* bb = ws + khalf * OUT_PAD + col;

    v8f c = {};
#pragma unroll
    for (int k = 0; k < HID_DIM; k += 4) {
        v2f a, b;
        a.x = aa[k];
        a.y = aa[k + 1];
        b.x = bb[k * OUT_PAD];
        b.y = bb[k * OUT_PAD + OUT_PAD];
        c = __builtin_amdgcn_wmma_f32_16x16x4_f32(false, a, false, b,
                                                  (short)0, c, false, false);
    }

    const int mrow = mbase + ((lane >> 4) << 3);
    if (col < OUT_DIM) {
#pragma unroll
        for (int r = 0; r < 8; ++r) {
            int m = mrow + r;
            if (m < n) H2[(size_t)m * OUT_DIM + col] = c[r];
        }
    }
}

// ---------------- aggregation layer 2 (OUT_DIM features, into d_out) ----------------

__global__ void k_agg2_init(const float* __restrict__ h2, const float* __restrict__ dinv,
                            float* __restrict__ out, int total) {
    int t = blockIdx.x * blockDim.x + threadIdx.x;
    if (t < total) {
        int i = t / OUT_DIM;
        float d = dinv[i];
        out[t] = d * d * h2[t];
    }
}

__global__ void k_agg2_edges(const long long* __restrict__ ei, const float* __restrict__ dinv,
                             const float* __restrict__ h2, float* __restrict__ out, int e) {
    int t = blockIdx.x * blockDim.x + threadIdx.x;
    int total = e * OUT_DIM;
    if (t >= total) return;
    int eg = t / OUT_DIM, f = t - eg * OUT_DIM;
    int r = (int)ei[eg];
    int c = (int)ei[(size_t)e + eg];
    float nrm = dinv[r] * dinv[c];
    atomicAdd(&out[(size_t)c * OUT_DIM + f], nrm * h2[(size_t)r * OUT_DIM + f]);
}

__global__ void k_log_softmax(float* __restrict__ out, const float* __restrict__ b2, int n) {
    int i = blockIdx.x * blockDim.x + threadIdx.x;
    if (i >= n) return;
    float v[OUT_DIM];
    float m = -INFINITY;
#pragma unroll
    for (int j = 0; j < OUT_DIM; ++j) {
        v[j] = out[(size_t)i * OUT_DIM + j] + b2[j];
        m = fmaxf(m, v[j]);
    }
    float s = 0.0f;
#pragma unroll
    for (int j = 0; j < OUT_DIM; ++j) s += __expf(v[j] - m);
    float l = m + __logf(s);
#pragma unroll
    for (int j = 0; j < OUT_DIM; ++j) out[(size_t)i * OUT_DIM + j] = v[j] - l;
}

// ---------------- launch ----------------

extern "C" void kernel_launch(void* const* d_in, const int* in_sizes, int n_in,
                              void* d_out, int out_size, void* d_ws, size_t ws_size,
                              hipStream_t stream) {
    const float*     x  = (const float*)d_in[0];
    const long long* ei = (const long long*)d_in[1];   // int64, shape (2, E)
    const float*     W1 = (const float*)d_in[2];
    const float*     b1 = (const float*)d_in[3];
    const float*     W2 = (const float*)d_in[4];
    const float*     b2 = (const float*)d_in[5];
    float*           out = (float*)d_out;

    const int n = in_sizes[0] / IN_DIM;   // 100000
    const int e = in_sizes[1] / 2;        // 1600000

    // workspace layout
    float* dinv = (float*)d_ws;                           // n
    float* h1   = dinv + n;                               // n * 64
    float* agg1 = h1 + (size_t)n * HID_DIM;               // n * 64
    float* h2   = agg1 + (size_t)n * HID_DIM;             // n * 40

    const int B = 256;
    // 1) degree -> rsqrt
    k_deg_init  <<<(n + B - 1) / B, B, 0, stream>>>(dinv, n);
    k_deg_accum <<<(e + B - 1) / B, B, 0, stream>>>(ei, dinv, e);
    k_deg_finish<<<(n + B - 1) / B, B, 0, stream>>>(dinv, n);

    // 2) GEMM1 (WMMA f32 16x16x4)
    const int mtiles = (n + 15) / 16;
    k_gemm1<<<mtiles, 128, 0, stream>>>(x, W1, h1, n);

    // 3) aggregate + SELU
    int tot1 = n * HID_DIM;
    k_agg1_init <<<(tot1 + B - 1) / B, B, 0, stream>>>(h1, dinv, agg1, tot1);
    int etot1 = e * HID_DIM;
    k_agg1_edges<<<(etot1 + B - 1) / B, B, 0, stream>>>(ei, dinv, h1, agg1, e);
    k_selu_bias <<<(tot1 + B - 1) / B, B, 0, stream>>>(agg1, b1, tot1);

    // 4) GEMM2 (WMMA f32 16x16x4, LDS-padded W2)
    k_gemm2<<<mtiles, 96, 0, stream>>>(agg1, W2, h2, n);

    // 5) aggregate into d_out, then log_softmax in place
    int tot2 = n * OUT_DIM;
    k_agg2_init <<<(tot2 + B - 1) / B, B, 0, stream>>>(h2, dinv, out, tot2);
    int etot2 = e * OUT_DIM;
    k_agg2_edges<<<(etot2 + B - 1) / B, B, 0, stream>>>(ei, dinv, h2, out, e);
    k_log_softmax<<<(n + B - 1) / B, B, 0, stream>>>(out, b2, n);
}